// Attention_82798379532303
// MI455X (gfx1250) — compile-verified
//
#include <hip/hip_runtime.h>
#include <cstdint>

typedef __attribute__((ext_vector_type(2))) float v2f;
typedef __attribute__((ext_vector_type(8))) float v8f;

#define NEG_INF_F (-1e20f)

constexpr int Bb = 32;
constexpr int Tt = 4096;
constexpr int Dd = 1024;
constexpr int S_CHUNKS = 16;              // split of T across workgroups
constexpr int CHUNK_T  = Tt / S_CHUNKS;   // 256 timesteps per workgroup
constexpr int SUB_T    = 16;              // WMMA M dimension (timesteps per sub-tile)
constexpr int ROWP     = Dd + 4;          // padded LDS row -> conflict-free WMMA A reads
constexpr int TILE_F   = SUB_T * ROWP;    // floats per LDS tile buffer (16448)

// dynamic LDS: tile[2][TILE_F] | cond[Dd] | wavePartials[32][16] | score[16]
constexpr int    SMEM_FLOATS = 2 * TILE_F + Dd + 32 * SUB_T + SUB_T;
constexpr size_t SMEM_BYTES  = (size_t)SMEM_FLOATS * sizeof(float);

// ---------------------------------------------------------------------------
// CDNA5 async global->LDS copy (ASYNCcnt-tracked). One call moves 64 B for
// this lane (4 x b128); the same immediate offset is applied to both the
// global and the LDS address, so the four transfers stay row-local.
// ---------------------------------------------------------------------------
__device__ __forceinline__ void async_copy_64B(const float* gptr, uint32_t ldsOff) {
    asm volatile(
        "global_load_async_to_lds_b128 %0, %1, off\n\t"
        "global_load_async_to_lds_b128 %0, %1, off offset:16\n\t"
        "global_load_async_to_lds_b128 %0, %1, off offset:32\n\t"
        "global_load_async_to_lds_b128 %0, %1, off offset:48"
        :: "v"(ldsOff), "v"(gptr) : "memory");
}

__device__ __forceinline__ void wait_async_zero() {
    asm volatile("s_wait_asynccnt 0x0" ::: "memory");
}

// ---------------------------------------------------------------------------
// Kernel 1: per-(batch, T-chunk) online-softmax partials.
//   grid = (S_CHUNKS, B), block = 1024 (32 waves). Thread tid owns d = tid.
//   Sub-tile pipeline: async-copy tile i+1 into LDS while computing tile i.
//   Scores: V_WMMA_F32_16X16X4_F32, wave w reduces K in [32w,32w+32) with
//   8 chained WMMAs fed from LDS; cross-wave reduction through LDS.
// ---------------------------------------------------------------------------
__global__ __launch_bounds__(1024)
void attn_chunk_kernel(const float* __restrict__ seq,
                       const long long* __restrict__ lens,
                       const float* __restrict__ cond,
                       float* __restrict__ ws_m,
                       float* __restrict__ ws_l,
                       float* __restrict__ ws_acc)
{
    const int c    = blockIdx.x;         // chunk index
    const int b    = blockIdx.y;         // batch index
    const int tid  = threadIdx.x;        // 0..1023, owns d = tid
    const int wave = tid >> 5;           // 0..31
    const int lane = tid & 31;

    const long long len = lens[b];
    const int  t_begin  = c * CHUNK_T;
    const int  pidx     = b * S_CHUNKS + c;
    float* accOut = ws_acc + (size_t)pidx * Dd;

    if ((long long)t_begin >= len) {     // fully masked chunk: zero partial
        accOut[tid] = 0.0f;
        if (tid == 0) { ws_m[pidx] = NEG_INF_F; ws_l[pidx] = 0.0f; }
        return;
    }

    extern __shared__ float smem[];
    float* tileBuf = smem;                       // [2][TILE_F]
    float* sCond   = smem + 2 * TILE_F;          // [Dd]
    float* sWave   = sCond + Dd;                 // [32][SUB_T]
    float* sScore  = sWave + 32 * SUB_T;         // [SUB_T]

    // Raw LDS byte offset of smem (flat-address low 32 bits == LDS offset).
    const uint32_t ldsBase = (uint32_t)(uintptr_t)(void*)smem;

    const float* seqB = seq + (size_t)b * Tt * Dd;
    sCond[tid] = cond[b * Dd + tid];             // visible after 1st barrier

    // Async-copy mapping: 1024 threads x 64 B cover one 16 x 1024 f32 tile.
    const int arow = tid >> 6;                   // 0..15 tile row
    const int acol = (tid & 63) << 4;            // float col: 16-float chunks
    const uint32_t ldsTileLane =
        (uint32_t)(arow * ROWP + acol) * 4u;     // byte offset inside a buffer

    const int t_end = (int)((len < (long long)(t_begin + CHUNK_T))
                                ? len : (long long)(t_begin + CHUNK_T));
    const int nTiles = (t_end - t_begin + SUB_T - 1) / SUB_T;   // >= 1

    // Prime the pipeline: tile 0 -> buffer 0.
    async_copy_64B(seqB + (size_t)(t_begin + arow) * Dd + acol,
                   ldsBase + ldsTileLane);

    float m = NEG_INF_F, l = 0.0f, acc = 0.0f;

    // WMMA A-operand lane mapping for 16x4 f32:
    //   lanes 0-15 hold (K=0,K=1) of row M=lane; lanes 16-31 hold (K=2,K=3).
    const int rowSel = lane >> 4;
    const int mrow   = lane & 15;
    const int k_base = wave * 32;                // this wave's K slice of D

    for (int i = 0; i < nTiles; ++i) {
        const int t0 = t_begin + i * SUB_T;
        const float* tileCur = tileBuf + (size_t)(i & 1) * TILE_F;

        wait_async_zero();       // this wave's copies for tile i complete
        __syncthreads();         // whole tile visible; all waves done with i-1

        if (i + 1 < nTiles) {    // stream tile i+1 into the other buffer
            const uint32_t off = ldsBase + (uint32_t)(((i + 1) & 1) * TILE_F * 4)
                               + ldsTileLane;
            async_copy_64B(seqB + (size_t)(t0 + SUB_T + arow) * Dd + acol, off);
        }

        // ---- score phase: C += A(seq tile, LDS) x B(cond replicated) -----
        v8f cm = {};
        const float* aBase = tileCur + mrow * ROWP + k_base + rowSel * 2;
        #pragma unroll
        for (int j = 0; j < 8; ++j) {
            v2f a, bv;
            a.x  = aBase[4 * j];
            a.y  = aBase[4 * j + 1];
            bv.x = sCond[k_base + 4 * j + rowSel * 2];
            bv.y = sCond[k_base + 4 * j + rowSel * 2 + 1];
            cm = __builtin_amdgcn_wmma_f32_16x16x4_f32(
                     false, a, false, bv, (short)0, cm, false, false);
        }
        // Column N=0 of C holds the 16 partial scores.
        if (lane == 0) {
            #pragma unroll
            for (int r = 0; r < 8; ++r) sWave[wave * SUB_T + r] = cm[r];
        } else if (lane == 16) {
            #pragma unroll
            for (int r = 0; r < 8; ++r) sWave[wave * SUB_T + 8 + r] = cm[r];
        }
        __syncthreads();

        // ---- cross-wave reduction + length mask --------------------------
        if (tid < SUB_T) {
            float s = 0.0f;
            #pragma unroll
            for (int w = 0; w < 32; ++w) s += sWave[w * SUB_T + tid];
            sScore[tid] = ((long long)(t0 + tid) < len) ? s : NEG_INF_F;
        }
        __syncthreads();

        // ---- online-softmax update (redundant scalar math per thread) ----
        float tmax = sScore[0];
        #pragma unroll
        for (int t = 1; t < SUB_T; ++t) tmax = fmaxf(tmax, sScore[t]);
        const float m_new = fmaxf(m, tmax);
        const float scale = __expf(m - m_new);   // 0 on first tile
        acc *= scale;
        l   *= scale;

        const float* col = tileCur + tid;        // LDS column d = tid
        #pragma unroll
        for (int t = 0; t < SUB_T; ++t) {
            const float w = __expf(sScore[t] - m_new);   // 0 for masked t
            l += w;
            acc = fmaf(w, col[t * ROWP], acc);
        }
        m = m_new;
        // next iteration's wait+barrier protects buffers, sWave and sScore
    }

    accOut[tid] = acc;
    if (tid == 0) { ws_m[pidx] = m; ws_l[pidx] = l; }
}

// ---------------------------------------------------------------------------
// Kernel 2: combine the S_CHUNKS partials per batch (log-sum-exp merge).
// ---------------------------------------------------------------------------
__global__ __launch_bounds__(1024)
void attn_combine_kernel(const float* __restrict__ ws_m,
                         const float* __restrict__ ws_l,
                         const float* __restrict__ ws_acc,
                         float* __restrict__ out)
{
    const int b   = blockIdx.x;
    const int tid = threadIdx.x;

    __shared__ float sM[S_CHUNKS];
    __shared__ float sL[S_CHUNKS];
    if (tid < S_CHUNKS) {
        sM[tid] = ws_m[b * S_CHUNKS + tid];
        sL[tid] = ws_l[b * S_CHUNKS + tid];
    }
    __syncthreads();

    float M = sM[0];
    #pragma unroll
    for (int i = 1; i < S_CHUNKS; ++i) M = fmaxf(M, sM[i]);

    float den = 0.0f, num = 0.0f;
    #pragma unroll
    for (int i = 0; i < S_CHUNKS; ++i) {
        const float f = __expf(sM[i] - M);       // 0 for fully-masked chunks
        den += f * sL[i];
        num = fmaf(f, ws_acc[((size_t)b * S_CHUNKS + i) * Dd + tid], num);
    }
    out[b * Dd + tid] = num / den;
}

// ---------------------------------------------------------------------------
extern "C" void kernel_launch(void* const* d_in, const int* in_sizes, int n_in,
                              void* d_out, int out_size, void* d_ws, size_t ws_size,
                              hipStream_t stream)
{
    const float*     seq  = (const float*)d_in[0];      // [B,T,D] f32
    const long long* lens = (const long long*)d_in[1];  // [B] int64
    const float*     cond = (const float*)d_in[2];      // [B,D] f32
    float* out = (float*)d_out;                         // [B,D] f32

    // Workspace layout: m[B*S] | l[B*S] | acc[B*S*D]   (~2.1 MB of floats)
    float* ws_m   = (float*)d_ws;
    float* ws_l   = ws_m + Bb * S_CHUNKS;
    float* ws_acc = ws_l + Bb * S_CHUNKS;

    // ~135 KB dynamic LDS (gfx1250 WGP has 320 KB) — raise the cap.
    (void)hipFuncSetAttribute((const void*)attn_chunk_kernel,
                              hipFuncAttributeMaxDynamicSharedMemorySize,
                              (int)SMEM_BYTES);

    attn_chunk_kernel<<<dim3(S_CHUNKS, Bb), 1024, SMEM_BYTES, stream>>>(
        seq, lens, cond, ws_m, ws_l, ws_acc);
    attn_combine_kernel<<<Bb, 1024, 0, stream>>>(ws_m, ws_l, ws_acc, out);
}